// ITSELF_7387343749583
// MI455X (gfx1250) — compile-verified
//
#include <hip/hip_runtime.h>

// ---------------------------------------------------------------------------
// Attention-rollout for gfx1250 (MI455X).
// Only layers 4 and 11 contribute. Pipeline per layer:
//   head-mean -> radix-select zeroing of k smallest -> (x+I)/2 + row-norm
//   -> result = attn @ result   (fp32 WMMA 16x16x4)
// All matrices padded to 208 (=13*16) so the WMMA GEMM has no edge cases.
// ---------------------------------------------------------------------------

typedef float v2f __attribute__((ext_vector_type(2)));
typedef float v8f __attribute__((ext_vector_type(8)));

#define NTOK   197
#define PAD    208            // 13 * 16
#define BATCH  32
#define HEADS  12
#define KSEL   9702u          // int(197*197*0.25)

// ---------------------------------------------------------------- identity --
__global__ void init_identity_kernel(float* __restrict__ res) {
    int idx = blockIdx.x * blockDim.x + threadIdx.x;
    const int per = PAD * PAD;
    if (idx >= BATCH * per) return;
    int r = idx % per;
    int i = r / PAD, j = r % PAD;
    res[idx] = (i == j && i < NTOK) ? 1.0f : 0.0f;
}

// --------------------------------------------------------------- head mean --
// attnLayer: (B, H, N, N) slice of the full tensor. mean: (B, PAD, PAD), pad=0.
__global__ void head_mean_kernel(const float* __restrict__ attnLayer,
                                 float* __restrict__ mean) {
    int idx = blockIdx.x * blockDim.x + threadIdx.x;
    const int per = PAD * PAD;
    if (idx >= BATCH * per) return;
    int b = idx / per;
    int r = idx % per;
    int i = r / PAD, j = r % PAD;
    float v = 0.0f;
    if (i < NTOK && j < NTOK) {
        const float* p = attnLayer + (size_t)b * HEADS * NTOK * NTOK
                                   + (size_t)i * NTOK + j;
        float s = 0.0f;
#pragma unroll
        for (int h = 0; h < HEADS; ++h) s += p[(size_t)h * NTOK * NTOK];
        v = s * (1.0f / HEADS);
    }
    mean[idx] = v;
}

// ------------------------------------------------- radix select + row norm --
// One workgroup per batch. Values are non-negative, so the IEEE bit pattern
// is order-preserving: 3-pass radix select (12 / 12 / 8 bits) finds the exact
// k-th smallest bit pattern; ties are zeroed in flat-index order like
// jax.lax.top_k. flat[:,0] is saved and restored. Then fused (x+I)*0.5 and
// row normalization (wave32 shuffle reduction).
__global__ void __launch_bounds__(512)
select_norm_kernel(float* __restrict__ mean) {
    __shared__ unsigned hist[4096];
    __shared__ unsigned eqIdx[2048];
    __shared__ unsigned sh_sel, sh_kRem, sh_eqCnt;
    __shared__ float    sh_v00;

    const int b = blockIdx.x;
    float* M = mean + (size_t)b * PAD * PAD;
    const int NN   = NTOK * NTOK;
    const int tid  = threadIdx.x;
    const int nthr = blockDim.x;

    // ---- pass 1: bits [31:20] ----
    for (int i = tid; i < 4096; i += nthr) hist[i] = 0;
    __syncthreads();
    for (int e = tid; e < NN; e += nthr) {
        int i = e / NTOK, j = e - i * NTOK;
        unsigned u = __float_as_uint(M[i * PAD + j]);
        atomicAdd(&hist[u >> 20], 1u);
    }
    __syncthreads();
    if (tid == 0) {
        unsigned cum = 0;
        for (int bin = 0; bin < 4096; ++bin) {
            unsigned h = hist[bin];
            if (cum + h >= KSEL) { sh_sel = (unsigned)bin; sh_kRem = KSEL - cum; break; }
            cum += h;
        }
    }
    __syncthreads();
    const unsigned b1 = sh_sel;
    const unsigned kRem1 = sh_kRem;

    // ---- pass 2: bits [19:8] among matching ----
    for (int i = tid; i < 4096; i += nthr) hist[i] = 0;
    __syncthreads();
    for (int e = tid; e < NN; e += nthr) {
        int i = e / NTOK, j = e - i * NTOK;
        unsigned u = __float_as_uint(M[i * PAD + j]);
        if ((u >> 20) == b1) atomicAdd(&hist[(u >> 8) & 0xFFFu], 1u);
    }
    __syncthreads();
    if (tid == 0) {
        unsigned cum = 0;
        for (int bin = 0; bin < 4096; ++bin) {
            unsigned h = hist[bin];
            if (cum + h >= kRem1) { sh_sel = (unsigned)bin; sh_kRem = kRem1 - cum; break; }
            cum += h;
        }
    }
    __syncthreads();
    const unsigned b2 = sh_sel;
    const unsigned kRem2 = sh_kRem;
    const unsigned pre24 = (b1 << 12) | b2;

    // ---- pass 3: bits [7:0] among matching ----
    for (int i = tid; i < 256; i += nthr) hist[i] = 0;
    __syncthreads();
    for (int e = tid; e < NN; e += nthr) {
        int i = e / NTOK, j = e - i * NTOK;
        unsigned u = __float_as_uint(M[i * PAD + j]);
        if ((u >> 8) == pre24) atomicAdd(&hist[u & 0xFFu], 1u);
    }
    __syncthreads();
    if (tid == 0) {
        unsigned cum = 0;
        for (int bin = 0; bin < 256; ++bin) {
            unsigned h = hist[bin];
            if (cum + h >= kRem2) { sh_sel = (unsigned)bin; sh_kRem = kRem2 - cum; break; }
            cum += h;
        }
        sh_v00  = M[0];
        sh_eqCnt = 0;
    }
    __syncthreads();
    const unsigned thr = (pre24 << 8) | sh_sel;   // bit pattern of k-th smallest
    const unsigned kEq = sh_kRem;                 // how many equals to zero

    // ---- zero everything strictly below thr; collect equals ----
    for (int e = tid; e < NN; e += nthr) {
        int i = e / NTOK, j = e - i * NTOK;
        int addr = i * PAD + j;
        unsigned u = __float_as_uint(M[addr]);
        if (u < thr) {
            M[addr] = 0.0f;
        } else if (u == thr) {
            unsigned p = atomicAdd(&sh_eqCnt, 1u);
            if (p < 2048u) eqIdx[p] = (unsigned)e;
        }
    }
    __syncthreads();
    unsigned ec = sh_eqCnt; if (ec > 2048u) ec = 2048u;
    for (unsigned p = tid; p < ec; p += nthr) {
        unsigned myE = eqIdx[p];
        unsigned rank = 0;
        for (unsigned q = 0; q < ec; ++q) rank += (eqIdx[q] < myE) ? 1u : 0u;
        if (rank < kEq) {
            int i = (int)(myE / NTOK), j = (int)(myE - (unsigned)i * NTOK);
            M[i * PAD + j] = 0.0f;
        }
    }
    __syncthreads();
    if (tid == 0) M[0] = sh_v00;   // restore flat[:,0] unconditionally
    __syncthreads();

    // ---- fused (x + I) * 0.5 and row normalization ----
    const int lane = tid & 31;
    const int wave = tid >> 5;
    const int nw   = nthr >> 5;
    for (int r = wave; r < NTOK; r += nw) {
        float part = 0.0f;
        for (int j = lane; j < NTOK; j += 32) {
            float x = M[r * PAD + j];
            x = 0.5f * (x + ((j == r) ? 1.0f : 0.0f));
            part += x;
        }
#pragma unroll
        for (int off = 16; off > 0; off >>= 1)
            part += __shfl_xor(part, off, 32);
        float inv = 1.0f / part;
        for (int j = lane; j < NTOK; j += 32) {
            float x = M[r * PAD + j];
            x = 0.5f * (x + ((j == r) ? 1.0f : 0.0f));
            M[r * PAD + j] = x * inv;
        }
    }
}

// ------------------------------------------------------------- WMMA GEMM ----
// O[b] = A[b] (PADxPAD) x R[b] (PADxPAD), fp32, V_WMMA_F32_16X16X4_F32.
// One wave per 16x16 output tile; K advances by 4 per WMMA.
// 32-bit A 16x4 layout : lanes 0-15 M rows, VGPR0/1 = K {0,1}; lanes 16-31 K {2,3}.
// 32-bit B 4x16 layout : VGPR0 = rows K0(l0-15)/K1(l16-31); VGPR1 = K2/K3.
// 32-bit C/D layout    : VGPR v, lanes 0-15 -> M=v, lanes 16-31 -> M=v+8.
__global__ void __launch_bounds__(128)
wmma_gemm_kernel(const float* __restrict__ A, const float* __restrict__ R,
                 float* __restrict__ O) {
    const int wave = threadIdx.x >> 5;
    const int lane = threadIdx.x & 31;
    const int tn = blockIdx.x * 4 + wave;   // tile col, valid < 13
    const int tm = blockIdx.y;              // tile row
    const int b  = blockIdx.z;
    if (tn >= PAD / 16) return;

    const float* Ab = A + (size_t)b * PAD * PAD;
    const float* Rb = R + (size_t)b * PAD * PAD;
    float*       Ob = O + (size_t)b * PAD * PAD;

    const int half = lane >> 4;      // 0 or 1
    const int l16  = lane & 15;
    const int arow = tm * 16 + l16;
    const int bcol = tn * 16 + l16;

    v8f acc = {};
    for (int kk = 0; kk < PAD; kk += 4) {
        v2f af = *(const v2f*)(Ab + (size_t)arow * PAD + kk + 2 * half);
        v2f bf;
        bf.x = Rb[(size_t)(kk + half) * PAD + bcol];
        bf.y = Rb[(size_t)(kk + 2 + half) * PAD + bcol];
        if (kk + 4 < PAD)
            __builtin_prefetch(Rb + (size_t)(kk + 4 + half) * PAD + bcol, 0, 0);
        acc = __builtin_amdgcn_wmma_f32_16x16x4_f32(
            /*neg_a=*/false, af, /*neg_b=*/false, bf,
            /*c_mod=*/(short)0, acc, /*reuse_a=*/false, /*reuse_b=*/false);
    }
#pragma unroll
    for (int v = 0; v < 8; ++v)
        Ob[(size_t)(tm * 16 + v + 8 * half) * PAD + bcol] = acc[v];
}

// ---------------------------------------------------------------- extract ---
__global__ void extract_kernel(const float* __restrict__ res,
                               float* __restrict__ out) {
    int idx = blockIdx.x * blockDim.x + threadIdx.x;
    const int per = NTOK * NTOK;
    if (idx >= BATCH * per) return;
    int b = idx / per;
    int r = idx % per;
    int i = r / NTOK, j = r % NTOK;
    out[idx] = res[(size_t)b * PAD * PAD + (size_t)i * PAD + j];
}

// ----------------------------------------------------------------- launch ---
extern "C" void kernel_launch(void* const* d_in, const int* in_sizes, int n_in,
                              void* d_out, int out_size, void* d_ws, size_t ws_size,
                              hipStream_t stream) {
    (void)in_sizes; (void)n_in; (void)out_size; (void)ws_size;
    const float* attn = (const float*)d_in[0];    // (12, 32, 12, 197, 197) fp32
    float* out = (float*)d_out;                   // (32, 197, 197) fp32

    const size_t perPad = (size_t)BATCH * PAD * PAD;
    float* mean = (float*)d_ws;
    float* res0 = mean + perPad;
    float* res1 = res0 + perPad;

    const size_t layerStride = (size_t)BATCH * HEADS * NTOK * NTOK;
    const int totPad  = BATCH * PAD * PAD;
    const int totReal = BATCH * NTOK * NTOK;

    init_identity_kernel<<<(totPad + 255) / 256, 256, 0, stream>>>(res0);

    dim3 gemmGrid(4, PAD / 16, BATCH);   // 4 waves/block x 4 blocks cover 13 tile cols

    // ---- layer 4 ----
    head_mean_kernel<<<(totPad + 255) / 256, 256, 0, stream>>>(
        attn + 4 * layerStride, mean);
    select_norm_kernel<<<BATCH, 512, 0, stream>>>(mean);
    wmma_gemm_kernel<<<gemmGrid, 128, 0, stream>>>(mean, res0, res1);

    // ---- layer 11 ----
    head_mean_kernel<<<(totPad + 255) / 256, 256, 0, stream>>>(
        attn + 11 * layerStride, mean);
    select_norm_kernel<<<BATCH, 512, 0, stream>>>(mean);
    wmma_gemm_kernel<<<gemmGrid, 128, 0, stream>>>(mean, res1, res0);

    extract_kernel<<<(totReal + 255) / 256, 256, 0, stream>>>(res0, out);
}